// DynamicSSM_Dense_34033320853881
// MI455X (gfx1250) — compile-verified
//
#include <hip/hip_runtime.h>
#include <math.h>

// ---------------------------------------------------------------------------
// DynamicSSM dense frequency-response kernel for gfx1250 (MI455X).
// Per frequency bin k: out_k[32x128] = ((X_k * C) * inv(I - A z_k)) * B
// Complex64 throughout; fp32 WMMA (v_wmma_f32_16x16x4_f32) for all GEMMs,
// VALU complex Gauss-Jordan (in LDS) for the 128x128 inverse.
// Wave tiling: wave w owns N-tile w and BOTH M-tiles, so B-fragments
// (global C/B, LDS W^-1) are loaded once per k-step and reused for 2 tiles.
// ---------------------------------------------------------------------------

typedef float v2f __attribute__((ext_vector_type(2)));
typedef float v8f __attribute__((ext_vector_type(8)));

#define FREQ   128
#define NFREQ  1024
#define BATCH  32
#define LDW    130            // padded LDS stride (floats) for 128-wide planes
#define LDU    130            // padded LDS stride for 32-row planes

// LDS partition (float offsets)
#define OFF_WRE 0
#define OFF_WIM (OFF_WRE + FREQ * LDW)
#define OFF_XS  (OFF_WIM + FREQ * LDW)
#define OFF_URE (OFF_XS  + BATCH * LDU)
#define OFF_UIM (OFF_URE + BATCH * LDU)
#define OFF_VRE (OFF_UIM + BATCH * LDU)
#define OFF_VIM (OFF_VRE + BATCH * LDU)
#define SMEM_FLOATS (OFF_VIM + BATCH * LDU)
#define SMEM_BYTES  (SMEM_FLOATS * sizeof(float))   // ~211 KB  (<= 320 KB/WGP)

__device__ __forceinline__ v8f wmma4(v2f a, v2f b, v8f c) {
  // D(16x16,f32) = A(16x4,f32) * B(4x16,f32) + C
  return __builtin_amdgcn_wmma_f32_16x16x4_f32(
      /*neg_a=*/false, a, /*neg_b=*/false, b,
      /*c_mod=*/(short)0, c, /*reuse_a=*/false, /*reuse_b=*/false);
}

__global__ __launch_bounds__(256)
void ssm_freq_kernel(const float*  __restrict__ x,    // [32,1024,128] f32
                     const float2* __restrict__ Ag,   // [128,128] c64
                     const float2* __restrict__ Bg,   // [128,128] c64
                     const float2* __restrict__ Cg,   // [128,128] c64
                     float2*       __restrict__ out)  // [32,1024,128] c64
{
  extern __shared__ float sm[];
  float* Wre = sm + OFF_WRE;
  float* Wim = sm + OFF_WIM;
  float* Xs  = sm + OFF_XS;
  float* Ure = sm + OFF_URE;
  float* Uim = sm + OFF_UIM;
  float* Vre = sm + OFF_VRE;
  float* Vim = sm + OFF_VIM;

  const int k   = blockIdx.x;
  const int tid = threadIdx.x;

  // Warm L2/WGP$ for the shared B/C matrices (128 KB each, fully L2-resident).
  if (tid < 128) {
    __builtin_prefetch((const char*)Cg + tid * 2048, 0, 1);
    __builtin_prefetch((const char*)Bg + tid * 2048, 0, 1);
  }

  // z_k = e^{-j 2 pi k / NFREQ}
  const float th = (float)k * (6.28318530717958647692f / (float)NFREQ);
  const float zr =  __cosf(th);
  const float zi = -__sinf(th);

  // ---- Build W = I - A*z_k (complex, split planes) and stage X tile --------
  for (int e = tid; e < FREQ * FREQ; e += 256) {
    const int i = e >> 7, j = e & 127;
    const float2 a = Ag[e];
    const float mr = a.x * zr - a.y * zi;
    const float mi = a.x * zi + a.y * zr;
    Wre[i * LDW + j] = (i == j ? 1.0f : 0.0f) - mr;
    Wim[i * LDW + j] = -mi;
  }
  for (int e = tid; e < BATCH * FREQ; e += 256) {
    const int b = e >> 7, i = e & 127;
    Xs[b * LDU + i] = x[((size_t)b * NFREQ + k) * FREQ + i];
  }
  __syncthreads();

  // ---- In-place complex Gauss-Jordan inverse (no pivoting) -----------------
  // diag(W) = 1 - A_ii z, |A_ii z| <= 0.5 => |diag| >= 0.5: pivot-safe.
  // 2 barriers per pivot: column-p fixup is folded into the elimination
  // sweep, and W[p][p]=1/d is written by a designated thread during
  // elimination (no thread reads W[p][p] in that phase).
  for (int p = 0; p < FREQ; ++p) {
    const float dr = Wre[p * LDW + p], di = Wim[p * LDW + p];
    const float s   = 1.0f / (dr * dr + di * di);
    const float idr = dr * s, idi = -di * s;          // 1/d
    if (tid < FREQ && tid != p) {                     // scale pivot row
      const float wr = Wre[p * LDW + tid], wi = Wim[p * LDW + tid];
      Wre[p * LDW + tid] = wr * idr - wi * idi;
      Wim[p * LDW + tid] = wr * idi + wi * idr;
    }
    __syncthreads();
    {                                                 // eliminate: 2 thr/row
      const int i  = tid >> 1;
      const int j0 = (tid & 1) * 64;
      if (i != p) {
        const float fr = Wre[i * LDW + p], fi = Wim[i * LDW + p];
        #pragma unroll 4
        for (int j = j0; j < j0 + 64; ++j) {
          if (j == p) {                               // folded column fixup
            Wre[i * LDW + p] = -(fr * idr - fi * idi);
            Wim[i * LDW + p] = -(fr * idi + fi * idr);
          } else {
            const float pr = Wre[p * LDW + j], pi = Wim[p * LDW + j];
            Wre[i * LDW + j] -= fr * pr - fi * pi;
            Wim[i * LDW + j] -= fr * pi + fi * pr;
          }
        }
      } else if ((tid & 1) == 0) {                    // row p: set pivot elt
        Wre[p * LDW + p] = idr;
        Wim[p * LDW + p] = idi;
      }
    }
    __syncthreads();
  }

  // ---- WMMA phase: wave w owns N-tile w, both M-tiles (batch 0-15 / 16-31) -
  const int lane = tid & 31;
  const int wv   = tid >> 5;       // 8 waves == 8 N-tiles
  const int lrow = lane & 15;      // M (for A frags) or N (for B and C/D frags)
  const int lhi  = lane >> 4;      // selects K pair {k0,k0+1} vs {k0+2,k0+3}
  const int col  = wv * 16 + lrow; // N coordinate for B/D fragments

  // Step A: U = X * C  (real A, complex B from global; B frags shared by mt)
  {
    v8f uRe0 = {}, uIm0 = {}, uRe1 = {}, uIm1 = {};
    for (int kk = 0; kk < FREQ; kk += 4) {
      const int krow = kk + 2 * lhi;
      v2f a0; a0.x = Xs[lrow * LDU + krow];
              a0.y = Xs[lrow * LDU + krow + 1];
      v2f a1; a1.x = Xs[(16 + lrow) * LDU + krow];
              a1.y = Xs[(16 + lrow) * LDU + krow + 1];
      const float2 c0 = Cg[(size_t)krow * FREQ + col];
      const float2 c1 = Cg[(size_t)(krow + 1) * FREQ + col];
      v2f bre; bre.x = c0.x; bre.y = c1.x;
      v2f bim; bim.x = c0.y; bim.y = c1.y;
      uRe0 = wmma4(a0, bre, uRe0);
      uIm0 = wmma4(a0, bim, uIm0);
      uRe1 = wmma4(a1, bre, uRe1);
      uIm1 = wmma4(a1, bim, uIm1);
    }
    #pragma unroll
    for (int r = 0; r < 8; ++r) {
      const int m = r + 8 * lhi;
      Ure[m * LDU + col]        = uRe0[r];
      Uim[m * LDU + col]        = uIm0[r];
      Ure[(16 + m) * LDU + col] = uRe1[r];
      Uim[(16 + m) * LDU + col] = uIm1[r];
    }
  }
  __syncthreads();

  // Step B: V = U * inv(W)  (complex x complex, both in LDS; W frags shared)
  {
    v8f re0 = {}, im0 = {}, re1 = {}, im1 = {};
    for (int kk = 0; kk < FREQ; kk += 4) {
      const int krow = kk + 2 * lhi;
      const int a0r  = lrow * LDU + krow;
      const int a1r  = (16 + lrow) * LDU + krow;
      v2f aR0; aR0.x = Ure[a0r]; aR0.y = Ure[a0r + 1];
      v2f aI0; aI0.x = Uim[a0r]; aI0.y = Uim[a0r + 1];
      v2f aN0; aN0.x = -aI0.x;   aN0.y = -aI0.y;      // fp32 WMMA: no A-neg
      v2f aR1; aR1.x = Ure[a1r]; aR1.y = Ure[a1r + 1];
      v2f aI1; aI1.x = Uim[a1r]; aI1.y = Uim[a1r + 1];
      v2f aN1; aN1.x = -aI1.x;   aN1.y = -aI1.y;
      v2f bR; bR.x = Wre[krow * LDW + col]; bR.y = Wre[(krow + 1) * LDW + col];
      v2f bI; bI.x = Wim[krow * LDW + col]; bI.y = Wim[(krow + 1) * LDW + col];
      re0 = wmma4(aR0, bR, re0);  re0 = wmma4(aN0, bI, re0);
      im0 = wmma4(aR0, bI, im0);  im0 = wmma4(aI0, bR, im0);
      re1 = wmma4(aR1, bR, re1);  re1 = wmma4(aN1, bI, re1);
      im1 = wmma4(aR1, bI, im1);  im1 = wmma4(aI1, bR, im1);
    }
    #pragma unroll
    for (int r = 0; r < 8; ++r) {
      const int m = r + 8 * lhi;
      Vre[m * LDU + col]        = re0[r];
      Vim[m * LDU + col]        = im0[r];
      Vre[(16 + m) * LDU + col] = re1[r];
      Vim[(16 + m) * LDU + col] = im1[r];
    }
  }
  __syncthreads();

  // Step C: out = V * B (B complex from global, frags shared), store complex64
  {
    v8f re0 = {}, im0 = {}, re1 = {}, im1 = {};
    for (int kk = 0; kk < FREQ; kk += 4) {
      const int krow = kk + 2 * lhi;
      const int a0r  = lrow * LDU + krow;
      const int a1r  = (16 + lrow) * LDU + krow;
      v2f aR0; aR0.x = Vre[a0r]; aR0.y = Vre[a0r + 1];
      v2f aI0; aI0.x = Vim[a0r]; aI0.y = Vim[a0r + 1];
      v2f aN0; aN0.x = -aI0.x;   aN0.y = -aI0.y;
      v2f aR1; aR1.x = Vre[a1r]; aR1.y = Vre[a1r + 1];
      v2f aI1; aI1.x = Vim[a1r]; aI1.y = Vim[a1r + 1];
      v2f aN1; aN1.x = -aI1.x;   aN1.y = -aI1.y;
      const float2 b0 = Bg[(size_t)krow * FREQ + col];
      const float2 b1 = Bg[(size_t)(krow + 1) * FREQ + col];
      v2f bR; bR.x = b0.x; bR.y = b1.x;
      v2f bI; bI.x = b0.y; bI.y = b1.y;
      re0 = wmma4(aR0, bR, re0);  re0 = wmma4(aN0, bI, re0);
      im0 = wmma4(aR0, bI, im0);  im0 = wmma4(aI0, bR, im0);
      re1 = wmma4(aR1, bR, re1);  re1 = wmma4(aN1, bI, re1);
      im1 = wmma4(aR1, bI, im1);  im1 = wmma4(aI1, bR, im1);
    }
    #pragma unroll
    for (int r = 0; r < 8; ++r) {
      const int m = r + 8 * lhi;                      // batch row (tile 0)
      float2 o0; o0.x = re0[r]; o0.y = im0[r];
      float2 o1; o1.x = re1[r]; o1.y = im1[r];
      out[((size_t)m * NFREQ + k) * FREQ + col]        = o0;
      out[((size_t)(16 + m) * NFREQ + k) * FREQ + col] = o1;
    }
  }
}

extern "C" void kernel_launch(void* const* d_in, const int* in_sizes, int n_in,
                              void* d_out, int out_size, void* d_ws, size_t ws_size,
                              hipStream_t stream) {
  (void)in_sizes; (void)n_in; (void)out_size; (void)d_ws; (void)ws_size;
  const float*  x  = (const float*) d_in[0];   // x_dynamic f32 [32,1024,128]
  const float2* Ag = (const float2*)d_in[1];   // A complex64 [128,128]
  const float2* Bg = (const float2*)d_in[2];   // B complex64 [128,128]
  const float2* Cg = (const float2*)d_in[3];   // C complex64 [128,128]
  float2*       o  = (float2*)      d_out;     // complex64 [32,1024,128]

  // >64KB dynamic LDS opt-in (no-op if already set; capture-safe, not a stream op)
  static bool attr_done = false;
  if (!attr_done) {
    hipFuncSetAttribute((const void*)ssm_freq_kernel,
                        hipFuncAttributeMaxDynamicSharedMemorySize,
                        (int)SMEM_BYTES);
    attr_done = true;
  }

  ssm_freq_kernel<<<NFREQ, 256, SMEM_BYTES, stream>>>(x, Ag, Bg, Cg, o);
}